// InfoNCELoss_13314398617870
// MI455X (gfx1250) — compile-verified
//
#include <hip/hip_runtime.h>
#include <math.h>

// CDNA5 / gfx1250: wave32, WMMA 16x16x4 fp32.
typedef __attribute__((ext_vector_type(2))) float v2f;
typedef __attribute__((ext_vector_type(8))) float v8f;

#define TEMP_INV 5.0f   // 1/0.2
#define DIM 256
#define NNEG 256

__device__ __forceinline__ float wave_sum(float v) {
    // wave32 butterfly reduction
    #pragma unroll
    for (int off = 16; off > 0; off >>= 1)
        v += __shfl_xor(v, off, 32);
    return v;
}

__global__ __launch_bounds__(256) void infonce_main(
    const float* __restrict__ U,          // [B, 256]
    const float* __restrict__ P,          // [B, 256]
    const float* __restrict__ Ng,         // [B, 256, 256]
    const unsigned char* __restrict__ Mk, // [B, 256] bool
    float* __restrict__ per_b)            // [B]
{
    const int b    = blockIdx.x;
    const int tid  = threadIdx.x;
    const int lane = tid & 31;
    const int wid  = tid >> 5;

    __shared__ float u_sh[DIM];
    __shared__ float dots[NNEG];
    __shared__ float rinv[NNEG];
    __shared__ float red3[24];
    __shared__ float red[8];
    __shared__ float s_uscale, s_pos, s_max;

    // ---- Phase A: load u, p; block-reduce |u|^2, |p|^2, u.p ----
    const float u = U[(size_t)b * DIM + tid];
    const float p = P[(size_t)b * DIM + tid];
    u_sh[tid] = u;

    float uu = wave_sum(u * u);
    float pp = wave_sum(p * p);
    float up = wave_sum(u * p);
    if (lane == 0) { red3[wid] = uu; red3[8 + wid] = pp; red3[16 + wid] = up; }
    __syncthreads();
    if (tid == 0) {
        float a = 0.f, c = 0.f, d = 0.f;
        for (int i = 0; i < 8; ++i) { a += red3[i]; c += red3[8 + i]; d += red3[16 + i]; }
        float un = fmaxf(sqrtf(a), 1e-12f);
        float pn = fmaxf(sqrtf(c), 1e-12f);
        s_uscale = TEMP_INV / un;                 // applied to raw dot(u, n_row)
        s_pos    = d / (un * pn) * TEMP_INV;      // pos_sim
    }
    __syncthreads();

    // ---- Phase B: WMMA f32 16x16x4 over negatives (2 tiles of 16 per wave) ----
    // A (16x4 fp32): lane l, half h=lane/16 supplies row (n0 + l%16), K = {2h, 2h+1}
    // B (4x16 fp32): u[d0 + 2h .. d0+2h+1] broadcast to all 16 columns
    const int h = lane >> 4;
    const int r = lane & 15;
    for (int t = 0; t < 2; ++t) {
        const int n0 = (wid * 2 + t) * 16;
        const float* rowp = Ng + (((size_t)b * NNEG) + (n0 + r)) * DIM + 2 * h;
        v8f c = {};
        float ssq = 0.f;
        for (int i = 0; i < 64; ++i) {
            v2f a = __builtin_nontemporal_load((const v2f*)(rowp + 4 * i));
            v2f bv = *(const v2f*)(&u_sh[4 * i + 2 * h]);
            c = __builtin_amdgcn_wmma_f32_16x16x4_f32(
                    /*neg_a=*/false, a, /*neg_b=*/false, bv,
                    /*c_mod=*/(short)0, c, /*reuse_a=*/false, /*reuse_b=*/false);
            ssq = fmaf(a.x, a.x, ssq);
            ssq = fmaf(a.y, a.y, ssq);
        }
        // combine half-row sums of squares: lane l + lane l+16 cover one row
        float full = ssq + __shfl_xor(ssq, 16, 32);
        if (h == 0) rinv[n0 + r] = 1.0f / fmaxf(sqrtf(full), 1e-12f);
        // D columns are identical; lane0 holds rows 0..7 in c[0..7], lane16 rows 8..15
        if (lane == 0) {
            #pragma unroll
            for (int k = 0; k < 8; ++k) dots[n0 + k] = c[k];
        }
        if (lane == 16) {
            #pragma unroll
            for (int k = 0; k < 8; ++k) dots[n0 + 8 + k] = c[k];
        }
    }
    __syncthreads();

    // ---- Phase C: masked logsumexp over [pos, 256 negatives] ----
    const float pos = s_pos;
    const unsigned char mk = Mk[(size_t)b * NNEG + tid];
    const float logit = mk ? dots[tid] * rinv[tid] * s_uscale : -1e30f;

    float m = fmaxf(logit, pos);
    #pragma unroll
    for (int off = 16; off > 0; off >>= 1) m = fmaxf(m, __shfl_xor(m, off, 32));
    if (lane == 0) red[wid] = m;
    __syncthreads();
    if (tid == 0) {
        float mm = red[0];
        for (int i = 1; i < 8; ++i) mm = fmaxf(mm, red[i]);
        s_max = mm;
    }
    __syncthreads();

    float e = (logit > -1e29f) ? expf(logit - s_max) : 0.0f;  // masked -> exact 0
    if (tid == 0) e += expf(pos - s_max);
    float s = wave_sum(e);
    __syncthreads();          // red[] reuse guard
    if (lane == 0) red[wid] = s;
    __syncthreads();
    if (tid == 0) {
        float ss = 0.f;
        for (int i = 0; i < 8; ++i) ss += red[i];
        per_b[b] = logf(ss) + s_max - pos;     // lse - pos_sim
    }
}

__global__ __launch_bounds__(256) void infonce_mean(
    const float* __restrict__ per_b, float* __restrict__ out, int B)
{
    __shared__ float red[8];
    const int tid  = threadIdx.x;
    const int lane = tid & 31;
    const int wid  = tid >> 5;
    float s = 0.f;
    for (int i = tid; i < B; i += 256) s += per_b[i];
    s = wave_sum(s);
    if (lane == 0) red[wid] = s;
    __syncthreads();
    if (tid == 0) {
        float ss = 0.f;
        for (int i = 0; i < 8; ++i) ss += red[i];
        out[0] = ss / (float)B;
    }
}

extern "C" void kernel_launch(void* const* d_in, const int* in_sizes, int n_in,
                              void* d_out, int out_size, void* d_ws, size_t ws_size,
                              hipStream_t stream) {
    const float* U  = (const float*)d_in[0];
    const float* P  = (const float*)d_in[1];
    const float* Ng = (const float*)d_in[2];
    const unsigned char* Mk = (const unsigned char*)d_in[3];

    const int B = in_sizes[0] / DIM;   // 2048
    float* per_b = (float*)d_ws;       // B floats of scratch
    float* out   = (float*)d_out;

    infonce_main<<<B, 256, 0, stream>>>(U, P, Ng, Mk, per_b);
    infonce_mean<<<1, 256, 0, stream>>>(per_b, out, B);
}